// LambdaRankLoss_8675833938015
// MI455X (gfx1250) — compile-verified
//
#include <hip/hip_runtime.h>
#include <math.h>

#define NITEMS 512
#define NROWS  64
#define SIGMA  1.0f
#define EPSF   1e-10f

typedef float v2f __attribute__((ext_vector_type(2)));
typedef float v8f __attribute__((ext_vector_type(8)));

// Wave32-wide sum via V_WMMA_F32_16X16X4_F32 with an all-ones B matrix.
// A holds one partial per lane in its first A-register (second = 0), so
// D[i][j] = sum_k A[i][k] for every column j (independent of A/B layout
// details because B is all ones).  Per the documented 16x16 f32 C/D layout,
// lane L (L<16) holds rows 0..7 of column L in D[0..7], lane L+16 holds rows
// 8..15 of column L.  Summing a lane's 8 D regs and xor-shuffling across the
// half-wave yields the full 32-lane sum in every lane.
__device__ __forceinline__ float wave_allsum_wmma(float x) {
  v2f a; a[0] = x;    a[1] = 0.0f;
  v2f b; b[0] = 1.0f; b[1] = 1.0f;
  v8f c = {};
  v8f d = __builtin_amdgcn_wmma_f32_16x16x4_f32(
      /*neg_a=*/false, a, /*neg_b=*/false, b,
      /*c_mod=*/(short)0, c, /*reuse_a=*/false, /*reuse_b=*/false);
  float s = d[0] + d[1] + d[2] + d[3] + d[4] + d[5] + d[6] + d[7];
  s += __shfl_xor(s, 16);
  return s;
}

// Block-wide sum (512 threads = 16 waves); every thread returns the total.
// Deterministic: fixed wave-reduce + fixed-order 16-element LDS sum.
__device__ __forceinline__ float block_allsum(float x, float* red) {
  float w = wave_allsum_wmma(x);
  int wave = threadIdx.x >> 5;
  __syncthreads();                      // protect red[] from prior use
  if ((threadIdx.x & 31) == 0) red[wave] = w;
  __syncthreads();
  float tot = 0.0f;
#pragma unroll
  for (int k = 0; k < 16; ++k) tot += red[k];
  return tot;
}

__global__ __launch_bounds__(NITEMS) void lambdarank_row_kernel(
    const float* __restrict__ scores, const float* __restrict__ rel,
    float* __restrict__ row_loss) {
  __shared__ float s_sc[NITEMS];
  __shared__ float s_rl[NITEMS];
  __shared__ float s_g[NITEMS];
  __shared__ float s_d[NITEMS];
  __shared__ int   hist[5];
  __shared__ float red[16];

  const int b = blockIdx.x;
  const int t = threadIdx.x;

  s_sc[t] = scores[b * NITEMS + t];
  s_rl[t] = rel[b * NITEMS + t];
  if (t < 5) hist[t] = 0;
  __syncthreads();

  // ---- Phase 1: 1-based rank under stable descending sort --------------
  // rank_i = 1 + #{j : s_j > s_i} + #{j < i : s_j == s_i}
  const float st = s_sc[t];
  int before = 0;
#pragma unroll 4
  for (int j = 0; j < NITEMS; ++j) {
    float sj = s_sc[j];                 // LDS broadcast (all lanes same j)
    before += (sj > st) || ((sj == st) && (j < t));
  }
  const float rankf = (float)(before + 1);
  const float rt    = s_rl[t];
  const float g     = exp2f(rt) - 1.0f;           // 2^rel - 1
  const float d     = 1.0f / log2f(rankf + 2.0f); // faithful: log2(rank+2)
  s_g[t] = g;
  s_d[t] = d;
  atomicAdd(&hist[(int)rt], 1);         // LDS histogram of rel in {0..4}
  __syncthreads();

  // ---- Phase 2: ideal DCG over the full list ---------------------------
  // Descending-sorted relevance reconstructed from the histogram: sorted
  // position t has relevance 4 while t<c4, then 3, etc.  gain(r)=2^r-1.
  const int c4 = hist[4], c3 = hist[3], c2 = hist[2], c1 = hist[1];
  float gain_p;
  if      (t < c4)                gain_p = 15.0f;
  else if (t < c4 + c3)           gain_p = 7.0f;
  else if (t < c4 + c3 + c2)      gain_p = 3.0f;
  else if (t < c4 + c3 + c2 + c1) gain_p = 1.0f;
  else                            gain_p = 0.0f;
  float idcg_part = gain_p / log2f((float)t + 2.0f);
  float idcg = block_allsum(idcg_part, red);
  idcg = fmaxf(idcg, EPSF);
  const float inv_idcg = 1.0f / idcg;

  // ---- Phase 3: pairwise loss over valid pairs (rel_i > rel_j) ---------
  float wsum = 0.0f, csum = 0.0f;
#pragma unroll 2
  for (int j = 0; j < NITEMS; ++j) {
    float rj = s_rl[j];
    if (rt > rj) {
      float dd    = (g - s_g[j]) * (d - s_d[j]);
      float delta = fabsf(dd) * inv_idcg;
      float sd    = st - s_sc[j];
      float pl    = log1pf(expf(-SIGMA * sd));   // |sd| <~ 10 => no overflow
      wsum += delta * pl;
      csum += 1.0f;
    }
  }
  float W = block_allsum(wsum, red);
  float C = block_allsum(csum, red);
  if (t == 0) row_loss[b] = W / (C + EPSF);
}

__global__ void lambdarank_finalize(const float* __restrict__ row_loss,
                                    float* __restrict__ out) {
  if (threadIdx.x == 0) {
    float s = 0.0f;
#pragma unroll
    for (int i = 0; i < NROWS; ++i) s += row_loss[i];
    out[0] = s * (1.0f / (float)NROWS);
  }
}

extern "C" void kernel_launch(void* const* d_in, const int* in_sizes, int n_in,
                              void* d_out, int out_size, void* d_ws, size_t ws_size,
                              hipStream_t stream) {
  (void)in_sizes; (void)n_in; (void)out_size; (void)ws_size;
  const float* scores = (const float*)d_in[0];
  const float* rel    = (const float*)d_in[1];
  float* out = (float*)d_out;
  float* ws  = (float*)d_ws;   // 64 per-row losses (256 B of scratch)

  lambdarank_row_kernel<<<NROWS, NITEMS, 0, stream>>>(scores, rel, ws);
  lambdarank_finalize<<<1, 32, 0, stream>>>(ws, out);
}